// BLIF_9388798509213
// MI455X (gfx1250) — compile-verified
//
#include <hip/hip_runtime.h>

typedef float v2f __attribute__((ext_vector_type(2)));
typedef float v8f __attribute__((ext_vector_type(8)));

#define T_LEN   256
#define B_SZ    8
#define C_CH    128
#define HW      196                 // 14*14
#define N_PIX   (B_SZ * C_CH * HW)  // 200704
#define N_BC    (B_SZ * C_CH)       // 1024
#define TILES_PER_BC 13             // ceil(196/16)
#define THRESH  1.0f

// One wave32 per 16-pixel tile of one (b,c) slice. The 16-step block scan
//   v_chunk = L * X_chunk + d^(i+1) * v_carry
// is computed with 4x V_WMMA_F32_16X16X4_F32 per chunk (fp32 end to end).
__global__ __launch_bounds__(256) void blif_scan_wmma(
    const float* __restrict__ x,      // [T, B, C, H, W]
    const float* __restrict__ A_log,  // [C]
    float* __restrict__ out)          // [T, B, C, H, W]
{
    const int wave = blockIdx.x * (blockDim.x >> 5) + (threadIdx.x >> 5);
    const int lane = threadIdx.x & 31;
    const int lo   = lane & 15;   // column (pixel) / row-within-half selector
    const int hi   = lane >> 4;   // half-wave selector

    const int bc   = wave / TILES_PER_BC;         // (b*128 + c)
    const int tile = wave - bc * TILES_PER_BC;
    const int c    = bc & (C_CH - 1);
    const int p0   = tile * 16;

    const float arate = expf(A_log[c]);           // a = exp(A_log)
    const float d     = expf(-arate);             // per-step decay, in [0.4, 0.9]

    // ---- A fragments: L[m][k] = (k<=m) ? d^(m-k) : 0, 16x4 chunks over K ----
    // 16x4 f32 A layout: VGPR0 = K=kb (lo) / K=kb+2 (hi), VGPR1 = K=kb+1 / kb+3
    v2f afrag[4];
#pragma unroll
    for (int q = 0; q < 4; ++q) {
        const int kb = 4 * q + 2 * hi;
        const int e0 = lo - kb;
        const int e1 = lo - kb - 1;
        afrag[q].x = (e0 >= 0) ? expf(-arate * (float)e0) : 0.0f;
        afrag[q].y = (e1 >= 0) ? expf(-arate * (float)e1) : 0.0f;
    }

    // ---- carry coefficients: row i = r + 8*hi gets d^(i+1) * v_carry ----
    float coef[8];
    {
        float cc = expf(-arate * (float)(8 * hi + 1));
#pragma unroll
        for (int r = 0; r < 8; ++r) { coef[r] = cc; cc *= d; }
    }

    // ---- per-lane pointers (tail tile: clamp loads, mask stores) ----
    const int  pvalid = (p0 + lo) < HW;
    const int  lo_cl  = pvalid ? lo : (HW - 1 - p0);
    const int  n0     = bc * HW + p0;

    // B layout: VGPR0 = row t0+4q+2*hi, VGPR1 = row t0+4q+1+2*hi, col = lo
    const float* px   = x   + n0 + lo_cl + 2 * hi * N_PIX;
    // C/D layout: VGPR r = row t0+r+8*hi, col = lo
    float*       pout = out + n0 + lo    + 8 * hi * N_PIX;

    float vprev = 0.0f;        // v at row 15 of previous chunk (per column)
    float sprev = 0.0f;        // pre-mask spike at row 15 of previous chunk

    for (int t0 = 0; t0 < T_LEN; t0 += 16) {
        // ---- load X chunk as four 4x16 B fragments (each b32 load = 2x64B) ----
        v2f bfrag[4];
#pragma unroll
        for (int q = 0; q < 4; ++q) {
            bfrag[q].x = px[(t0 + 4 * q)     * N_PIX];
            bfrag[q].y = px[(t0 + 4 * q + 1) * N_PIX];
        }

        // ---- v_local = L * X via 4 accumulating fp32 WMMAs ----
        v8f acc = {0.0f, 0.0f, 0.0f, 0.0f, 0.0f, 0.0f, 0.0f, 0.0f};
        acc = __builtin_amdgcn_wmma_f32_16x16x4_f32(false, afrag[0], false, bfrag[0], (short)0, acc, false, false);
        acc = __builtin_amdgcn_wmma_f32_16x16x4_f32(false, afrag[1], false, bfrag[1], (short)0, acc, false, false);
        acc = __builtin_amdgcn_wmma_f32_16x16x4_f32(false, afrag[2], false, bfrag[2], (short)0, acc, false, false);
        acc = __builtin_amdgcn_wmma_f32_16x16x4_f32(false, afrag[3], false, bfrag[3], (short)0, acc, false, false);

        // ---- add carry, threshold (pre-mask spikes) ----
        float sp[8];
#pragma unroll
        for (int r = 0; r < 8; ++r) {
            const float v = acc[r] + coef[r] * vprev;
            acc[r] = v;
            sp[r]  = (v > THRESH) ? 1.0f : 0.0f;
        }

        // cross-half / cross-chunk carries (rows 7 and 15 live in sp[7]/acc[7])
        const float sp7        = sp[7];
        const float cross      = __shfl(sp7,      lo,      32); // hi lanes: row 7 (lo half)
        const float sprev_next = __shfl(sp7,      lo + 16, 32); // row 15 (hi half)
        const float vprev_next = __shfl(acc[7],   lo + 16, 32); // v row 15

        // ---- refractory mask with PRE-mask previous spike ----
        float o[8];
        const float prev0 = hi ? cross : sprev;
        o[0] = sp[0] * (1.0f - prev0);
#pragma unroll
        for (int r = 1; r < 8; ++r) o[r] = sp[r] * (1.0f - sp[r - 1]);

        if (pvalid) {
#pragma unroll
            for (int r = 0; r < 8; ++r) pout[(t0 + r) * N_PIX] = o[r];
        }

        vprev = vprev_next;
        sprev = sprev_next;
    }
}

extern "C" void kernel_launch(void* const* d_in, const int* in_sizes, int n_in,
                              void* d_out, int out_size, void* d_ws, size_t ws_size,
                              hipStream_t stream) {
    const float* x     = (const float*)d_in[0];   // [256, 8, 128, 14, 14] f32
    const float* A_log = (const float*)d_in[1];   // [128] f32
    float*       out   = (float*)d_out;           // [256, 8, 128, 14, 14] f32
    (void)in_sizes; (void)n_in; (void)out_size; (void)d_ws; (void)ws_size;

    const int total_waves     = N_BC * TILES_PER_BC;   // 13312
    const int waves_per_block = 8;                     // 256 threads
    const int grid            = total_waves / waves_per_block; // 1664, exact

    blif_scan_wmma<<<grid, 256, 0, stream>>>(x, A_log, out);
}